// DP_DDGCN_28630251995701
// MI455X (gfx1250) — compile-verified
//
#include <hip/hip_runtime.h>
#include <math.h>

#define Bn 16
#define Nn 256
#define Hh 64
#define FP 6
#define TT 24
#define PL 12

typedef __attribute__((ext_vector_type(2))) float v2f;
typedef __attribute__((ext_vector_type(8))) float v8f;

// ---------------------------------------------------------------------------
// angle matrix: ang[i][j] = atan2(lat_j - lat_i, lon_j - lon_i)
// ---------------------------------------------------------------------------
__global__ void k_ang(const float* __restrict__ coords, float* __restrict__ ang) {
    int i = blockIdx.x, j = threadIdx.x;
    float dlon = coords[j * 2 + 0] - coords[i * 2 + 0];
    float dlat = coords[j * 2 + 1] - coords[i * 2 + 1];
    ang[i * Nn + j] = atan2f(dlat, dlon);
}

// ---------------------------------------------------------------------------
// dynamic adjacency: w_out[b,i,j] = adj[i,j] * max(spd_b*cos(ang[i,j]-rad_b),0)
// ---------------------------------------------------------------------------
__global__ void k_dyn(const float* __restrict__ adj, const float* __restrict__ ang,
                      const float* __restrict__ hw, int t, float* __restrict__ wout) {
    int i = blockIdx.x, b = blockIdx.y, j = threadIdx.x;
    float wd  = hw[(b * TT + t) * 4 + 1];
    float spd = hw[(b * TT + t) * 4 + 2];
    float rad = wd * 0.017453292519943295f;
    float wdyn = fmaxf(spd * cosf(ang[i * Nn + j] - rad), 0.f);
    wout[(b * Nn + i) * Nn + j] = adj[i * Nn + j] * wdyn;
}

// ---------------------------------------------------------------------------
// degree vectors for w_out (rowsum) and w_outT (colsum); +1 for identity diag
// ---------------------------------------------------------------------------
__global__ void k_sums(const float* __restrict__ wout, float* __restrict__ dov,
                       float* __restrict__ din) {
    int b = blockIdx.x, j = threadIdx.x;
    const float* W = wout + b * Nn * Nn;
    float rs = 0.f, cs = 0.f;
    for (int k = 0; k < Nn; ++k) { rs += W[j * Nn + k]; cs += W[k * Nn + j]; }
    dov[b * Nn + j] = 1.f / (sqrtf(rs + 1.f) + 1e-6f);
    din[b * Nn + j] = 1.f / (sqrtf(cs + 1.f) + 1e-6f);
}

// ---------------------------------------------------------------------------
// Y[b] = norm(w_out[b] (+transpose) + I) @ X[b]   via fp32 WMMA 16x16x4.
// Normalization d_i*(A+I)*d_k fused into the A-fragment; A loaded once per
// K-step and reused across TN N-tiles (TN accumulators per wave).
// Branchless column masking (clamped index * mask) -> no exec divergence in
// the inner loop.
// ---------------------------------------------------------------------------
template <int TN, int TRANS>
__global__ __launch_bounds__(32) void k_gnorm_t(
    const float* __restrict__ wout, const float* __restrict__ dvec,
    const float* __restrict__ X, int bsX, int ldx,
    float* __restrict__ Y, int bsY, int ldy, int F) {
    int tm = blockIdx.x, tn0 = blockIdx.y * TN, b = blockIdx.z;
    int lane = threadIdx.x, half = lane >> 4, l15 = lane & 15;
    const float* A  = wout + b * Nn * Nn;
    const float* d  = dvec + b * Nn;
    const float* Xb = X + b * bsX;
    int row = tm * 16 + l15;
    float di = d[row];

    int   colc[TN];
    float mc[TN];
#pragma unroll
    for (int i = 0; i < TN; ++i) {
        int col = (tn0 + i) * 16 + l15;
        colc[i] = col < F ? col : 0;
        mc[i]   = col < F ? 1.f : 0.f;
    }

    v8f acc[TN] = {};
    for (int k0 = 0; k0 < Nn; k0 += 4) {
        int ka = k0 + 2 * half;          // even -> v2f loads are 8B aligned
        int kb = ka + 1;
        v2f dp = *(const v2f*)(d + ka);
        v2f av;
        if (TRANS) {
            av.x = A[ka * Nn + row];
            av.y = A[kb * Nn + row];
        } else {
            av = *(const v2f*)(A + row * Nn + ka);
        }
        av.x = (av.x + (row == ka ? 1.f : 0.f)) * (di * dp.x);
        av.y = (av.y + (row == kb ? 1.f : 0.f)) * (di * dp.y);
#pragma unroll
        for (int i = 0; i < TN; ++i) {
            v2f bv;
            bv.x = Xb[ka * ldx + colc[i]] * mc[i];
            bv.y = Xb[kb * ldx + colc[i]] * mc[i];
            acc[i] = __builtin_amdgcn_wmma_f32_16x16x4_f32(false, av, false, bv,
                                                           (short)0, acc[i], false, false);
        }
    }
#pragma unroll
    for (int i = 0; i < TN; ++i) {
#pragma unroll
        for (int v = 0; v < 8; ++v) {
            int m = tm * 16 + v + 8 * half;
            int n = (tn0 + i) * 16 + l15;
            if (n < F) Y[b * bsY + m * ldy + n] = acc[i][v];
        }
    }
}

// ---------------------------------------------------------------------------
// Y = X (M x K) @ W^T (W is Nout x K row-major) + bias, optional ReLU.
// A fragment loaded once per K-step, reused across TN N-tiles.
// ---------------------------------------------------------------------------
template <int TN>
__global__ __launch_bounds__(32) void k_gemm_t(
    const float* __restrict__ X, int ldx,
    const float* __restrict__ W, const float* __restrict__ bias,
    float* __restrict__ Y, int ldy, int K, int Nout, int act) {
    int tm = blockIdx.x, tn0 = blockIdx.y * TN;
    int lane = threadIdx.x, half = lane >> 4, l15 = lane & 15;
    int row = tm * 16 + l15;

    int   colc[TN];
    float mc[TN];
#pragma unroll
    for (int i = 0; i < TN; ++i) {
        int col = (tn0 + i) * 16 + l15;
        colc[i] = col < Nout ? col : 0;
        mc[i]   = col < Nout ? 1.f : 0.f;
    }

    v8f acc[TN] = {};
    int Kpad = (K + 3) & ~3;
    for (int k0 = 0; k0 < Kpad; k0 += 4) {
        int ka = k0 + 2 * half, kb = ka + 1;
        int   kac = ka < K ? ka : 0, kbc = kb < K ? kb : 0;
        float ma  = ka < K ? 1.f : 0.f, mb = kb < K ? 1.f : 0.f;
        v2f av;
        av.x = X[row * ldx + kac] * ma;      // a==0 when k out of range -> bv irrelevant
        av.y = X[row * ldx + kbc] * mb;
#pragma unroll
        for (int i = 0; i < TN; ++i) {
            v2f bv;
            bv.x = W[colc[i] * K + kac] * mc[i];
            bv.y = W[colc[i] * K + kbc] * mc[i];
            acc[i] = __builtin_amdgcn_wmma_f32_16x16x4_f32(false, av, false, bv,
                                                           (short)0, acc[i], false, false);
        }
    }
#pragma unroll
    for (int i = 0; i < TN; ++i) {
#pragma unroll
        for (int v = 0; v < 8; ++v) {
            int m = tm * 16 + v + 8 * half;
            int n = (tn0 + i) * 16 + l15;
            if (n < Nout) {
                float y = acc[i][v] + bias[n];
                if (act) y = fmaxf(y, 0.f);
                Y[m * ldy + n] = y;
            }
        }
    }
}

// ---------------------------------------------------------------------------
// GRU gate math: h = (1-z)*n + z*h,  r/z = sigmoid, n = tanh(i_n + r*h_n)
// ---------------------------------------------------------------------------
__global__ void k_gru(const float* __restrict__ gi, const float* __restrict__ gh,
                      float* __restrict__ h) {
    int idx = blockIdx.x * blockDim.x + threadIdx.x;   // 4096*64
    int r = idx >> 6, c = idx & 63;
    float ir = gi[r * 192 + c], iz = gi[r * 192 + 64 + c], in_ = gi[r * 192 + 128 + c];
    float hr = gh[r * 192 + c], hz = gh[r * 192 + 64 + c], hn  = gh[r * 192 + 128 + c];
    float rr = 1.f / (1.f + expf(-(ir + hr)));
    float z  = 1.f / (1.f + expf(-(iz + hz)));
    float nn = tanhf(in_ + rr * hn);
    h[idx] = (1.f - z) * nn + z * h[idx];
}

__global__ void k_zero(float* __restrict__ p, int n) {
    int idx = blockIdx.x * blockDim.x + threadIdx.x;
    if (idx < n) p[idx] = 0.f;
}

// scatter prediction p into out[(b, p, i, f)]
__global__ void k_copy(const float* __restrict__ decx, float* __restrict__ out, int p) {
    int idx = blockIdx.x * blockDim.x + threadIdx.x;   // 4096*6
    int r = idx / 6, f = idx - r * 6;
    int b = r >> 8, i = r & 255;
    out[((b * PL + p) * Nn + i) * FP + f] = decx[idx];
}

// ---------------------------------------------------------------------------
extern "C" void kernel_launch(void* const* d_in, const int* in_sizes, int n_in,
                              void* d_out, int out_size, void* d_ws, size_t ws_size,
                              hipStream_t stream) {
    (void)in_sizes; (void)n_in; (void)out_size; (void)ws_size;
    const float* hp     = (const float*)d_in[0];
    const float* hw     = (const float*)d_in[1];
    const float* adj    = (const float*)d_in[2];
    const float* coords = (const float*)d_in[3];
    const float* w_gc1  = (const float*)d_in[4];
    const float* b_gc1  = (const float*)d_in[5];
    const float* w_gc2  = (const float*)d_in[6];
    const float* b_gc2  = (const float*)d_in[7];
    const float* w_ih   = (const float*)d_in[8];
    const float* w_hh   = (const float*)d_in[9];
    const float* b_ih   = (const float*)d_in[10];
    const float* b_hh   = (const float*)d_in[11];
    const float* w_m1   = (const float*)d_in[12];
    const float* b_m1   = (const float*)d_in[13];
    const float* w_m2   = (const float*)d_in[14];
    const float* b_m2   = (const float*)d_in[15];
    float* out = (float*)d_out;

    float* ws   = (float*)d_ws;
    float* ang  = ws;  ws += Nn * Nn;
    float* wout = ws;  ws += Bn * Nn * Nn;
    float* dov  = ws;  ws += Bn * Nn;
    float* din  = ws;  ws += Bn * Nn;
    float* y1   = ws;  ws += Bn * Nn * FP;
    float* h1   = ws;  ws += Bn * Nn * Hh;
    float* y2   = ws;  ws += Bn * Nn * Hh;
    float* sfb  = ws;  ws += Bn * Nn * Hh;
    float* gi   = ws;  ws += Bn * Nn * 3 * Hh;
    float* gh   = ws;  ws += Bn * Nn * 3 * Hh;
    float* hbuf = ws;  ws += Bn * Nn * Hh;
    float* m1b  = ws;  ws += Bn * Nn * 32;
    float* decx = ws;  ws += Bn * Nn * FP;

    const int M = Bn * Nn;                       // 4096 flattened rows

    k_ang<<<Nn, Nn, 0, stream>>>(coords, ang);
    k_zero<<<(M * Hh + 255) / 256, 256, 0, stream>>>(hbuf, M * Hh);

    // ---------------- encoder: 24 sequential steps ----------------
    for (int t = 0; t < TT; ++t) {
        k_dyn<<<dim3(Nn, Bn), Nn, 0, stream>>>(adj, ang, hw, t, wout);
        k_sums<<<Bn, Nn, 0, stream>>>(wout, dov, din);
        // gconv1: y1 = norm_out @ x_t ; h1 = relu(y1 @ w_gc1^T + b)
        k_gnorm_t<1, 0><<<dim3(16, 1, Bn), 32, 0, stream>>>(
            wout, dov, hp + t * Nn * FP, TT * Nn * FP, FP, y1, Nn * FP, FP, FP);
        k_gemm_t<4><<<dim3(M / 16, 1), 32, 0, stream>>>(y1, FP, w_gc1, b_gc1, h1, Hh, FP, Hh, 1);
        // gconv2: y2 = norm_in @ h1 ; sf = relu(y2 @ w_gc2^T + b)
        k_gnorm_t<4, 1><<<dim3(16, 1, Bn), 32, 0, stream>>>(
            wout, din, h1, Nn * Hh, Hh, y2, Nn * Hh, Hh, Hh);
        k_gemm_t<4><<<dim3(M / 16, 1), 32, 0, stream>>>(y2, Hh, w_gc2, b_gc2, sfb, Hh, Hh, Hh, 1);
        // GRU
        k_gemm_t<4><<<dim3(M / 16, 3), 32, 0, stream>>>(sfb, Hh, w_ih, b_ih, gi, 192, Hh, 192, 0);
        k_gemm_t<4><<<dim3(M / 16, 3), 32, 0, stream>>>(hbuf, Hh, w_hh, b_hh, gh, 192, Hh, 192, 0);
        k_gru<<<(M * Hh) / 256, 256, 0, stream>>>(gi, gh, hbuf);
    }

    // dec0 = mlp(h)
    k_gemm_t<2><<<dim3(M / 16, 1), 32, 0, stream>>>(hbuf, Hh, w_m1, b_m1, m1b, 32, Hh, 32, 1);
    k_gemm_t<1><<<dim3(M / 16, 1), 32, 0, stream>>>(m1b, 32, w_m2, b_m2, decx, FP, 32, FP, 0);

    // ---------------- decoder: 12 sequential steps (fixed last w_out) -------
    for (int p = 0; p < PL; ++p) {
        k_gnorm_t<1, 0><<<dim3(16, 1, Bn), 32, 0, stream>>>(
            wout, dov, decx, Nn * FP, FP, y1, Nn * FP, FP, FP);
        k_gemm_t<4><<<dim3(M / 16, 1), 32, 0, stream>>>(y1, FP, w_gc1, b_gc1, h1, Hh, FP, Hh, 1);
        k_gnorm_t<4, 1><<<dim3(16, 1, Bn), 32, 0, stream>>>(
            wout, din, h1, Nn * Hh, Hh, y2, Nn * Hh, Hh, Hh);
        k_gemm_t<4><<<dim3(M / 16, 1), 32, 0, stream>>>(y2, Hh, w_gc2, b_gc2, sfb, Hh, Hh, Hh, 1);
        k_gemm_t<4><<<dim3(M / 16, 3), 32, 0, stream>>>(sfb, Hh, w_ih, b_ih, gi, 192, Hh, 192, 0);
        k_gemm_t<4><<<dim3(M / 16, 3), 32, 0, stream>>>(hbuf, Hh, w_hh, b_hh, gh, 192, Hh, 192, 0);
        k_gru<<<(M * Hh) / 256, 256, 0, stream>>>(gi, gh, hbuf);
        k_gemm_t<2><<<dim3(M / 16, 1), 32, 0, stream>>>(hbuf, Hh, w_m1, b_m1, m1b, 32, Hh, 32, 1);
        k_gemm_t<1><<<dim3(M / 16, 1), 32, 0, stream>>>(m1b, 32, w_m2, b_m2, decx, FP, 32, FP, 0);
        k_copy<<<(M * FP + 255) / 256, 256, 0, stream>>>(decx, out, p);
    }
}